// PowerLinear_12180527251822
// MI455X (gfx1250) — compile-verified
//
#include <hip/hip_runtime.h>

typedef __attribute__((ext_vector_type(2))) float v2f;
typedef __attribute__((ext_vector_type(8))) float v8f;

#define DIM 1024
#define BATCH 64

// C[M,DIM] = A[M,DIM] @ B ; BT=false: B is [DIM,DIM] row-major;
// BT=true: B_eff[k][n] = B[n*DIM + k]  (i.e. multiply by B^T of the row-major array)
template <bool BT>
__global__ __launch_bounds__(128) void gemm_f32_wmma(const float* __restrict__ A,
                                                     const float* __restrict__ B,
                                                     float* __restrict__ C) {
    const int lane   = threadIdx.x & 31;
    const int wave   = threadIdx.x >> 5;
    const int tilesN = DIM >> 4;                    // 64
    const int tile   = blockIdx.x * 4 + wave;
    const int m0     = (tile / tilesN) << 4;
    const int n0     = (tile % tilesN) << 4;
    const int half   = lane >> 4;                   // 0: K pair {0,1}, 1: K pair {2,3}
    const int l      = lane & 15;
    const int kOff   = half << 1;

    // Streaming pointers: advance by constant stride; loads use immediate offsets.
    const float* aPtr = A + (m0 + l) * DIM + kOff;
    const float* bPtr = BT ? (B + (n0 + l) * DIM + kOff)
                           : (B + kOff * DIM + n0 + l);
    const ptrdiff_t bStep = BT ? 16 : 16 * DIM;

    v8f acc0 = {}, acc1 = {}, acc2 = {}, acc3 = {};

    for (int k0 = 0; k0 < DIM; k0 += 16) {
        v2f a0 = *(const v2f*)(aPtr + 0);
        v2f a1 = *(const v2f*)(aPtr + 4);
        v2f a2 = *(const v2f*)(aPtr + 8);
        v2f a3 = *(const v2f*)(aPtr + 12);
        aPtr += 16;

        v2f b0, b1, b2, b3;
        if (BT) {
            b0 = *(const v2f*)(bPtr + 0);
            b1 = *(const v2f*)(bPtr + 4);
            b2 = *(const v2f*)(bPtr + 8);
            b3 = *(const v2f*)(bPtr + 12);
        } else {
            b0.x = bPtr[0 * DIM];  b0.y = bPtr[1 * DIM];
            b1.x = bPtr[4 * DIM];  b1.y = bPtr[5 * DIM];
            b2.x = bPtr[8 * DIM];  b2.y = bPtr[9 * DIM];
            b3.x = bPtr[12 * DIM]; b3.y = bPtr[13 * DIM];
        }
        // Warm L2 ahead on the B stream (global_prefetch_b8, speculative: OOB is dropped).
        __builtin_prefetch(bPtr + 16 * bStep, 0, 1);
        bPtr += bStep;

        // D = A(16x4,f32) * B(4x16,f32) + C(16x16,f32): exact f32 accumulate (RNE)
        acc0 = __builtin_amdgcn_wmma_f32_16x16x4_f32(false, a0, false, b0,
                                                     (short)0, acc0, false, false);
        acc1 = __builtin_amdgcn_wmma_f32_16x16x4_f32(false, a1, false, b1,
                                                     (short)0, acc1, false, false);
        acc2 = __builtin_amdgcn_wmma_f32_16x16x4_f32(false, a2, false, b2,
                                                     (short)0, acc2, false, false);
        acc3 = __builtin_amdgcn_wmma_f32_16x16x4_f32(false, a3, false, b3,
                                                     (short)0, acc3, false, false);
    }

    union { v8f v; float f[8]; } out;
    out.v = (acc0 + acc1) + (acc2 + acc3);
    float* cPtr = C + (m0 + half * 8) * DIM + n0 + l;   // VGPR r -> rows r / r+8
#pragma unroll
    for (int r = 0; r < 8; ++r) cPtr[r * DIM] = out.f[r];
}

// Middle pass: T2[b,j] = (n_b==0) ? sum_k T[b,k] : T[b,j] * d_j^{n_b}
// (n_b==0 reproduces the elementwise 0^0=1 -> all-ones-matrix semantics of the reference)
__global__ __launch_bounds__(256) void apply_power(const float* __restrict__ T,
                                                   const int* __restrict__ n,
                                                   const float* __restrict__ diag,
                                                   float* __restrict__ T2) {
    __shared__ float red[256];
    const int b  = blockIdx.x;
    const int t  = threadIdx.x;
    const int nb = n[b];

    float v[4];
    float local = 0.f;
#pragma unroll
    for (int i = 0; i < 4; ++i) {
        v[i] = T[b * DIM + t + i * 256];
        local += v[i];
    }
    red[t] = local;
    __syncthreads();
#pragma unroll
    for (int s = 128; s > 0; s >>= 1) {
        if (t < s) red[t] += red[t + s];
        __syncthreads();
    }
    const float rowsum = red[0];

#pragma unroll
    for (int i = 0; i < 4; ++i) {
        const int j = t + i * 256;
        float outv;
        if (nb == 0) {
            outv = rowsum;
        } else {
            const float d = diag[j];
            float p = d;
            for (int q = 1; q < nb; ++q) p *= d;   // n in [0,6)
            outv = v[i] * p;
        }
        T2[b * DIM + j] = outv;
    }
}

extern "C" void kernel_launch(void* const* d_in, const int* in_sizes, int n_in,
                              void* d_out, int out_size, void* d_ws, size_t ws_size,
                              hipStream_t stream) {
    (void)in_sizes; (void)n_in; (void)out_size; (void)ws_size;
    const float* x    = (const float*)d_in[0];   // (64,1024) f32
    const int*   nvec = (const int*)d_in[1];     // (64,) i32
    const float* diag = (const float*)d_in[2];   // (1024,) f32
    const float* rot  = (const float*)d_in[3];   // (1024,1024) f32 row-major

    float* y  = (float*)d_out;                   // (64,1024) f32
    float* T  = (float*)d_ws;                    // 64*1024 f32
    float* T2 = T + BATCH * DIM;                 // 64*1024 f32

    const dim3 gblock(128);
    const dim3 ggrid(((BATCH / 16) * (DIM / 16)) / 4);  // 256 tiles, 4 waves/block

    // T = X @ R
    gemm_f32_wmma<false><<<ggrid, gblock, 0, stream>>>(x, rot, T);
    // T2 = diag-power / row-sum middle pass
    apply_power<<<dim3(BATCH), dim3(256), 0, stream>>>(T, nvec, diag, T2);
    // Y = T2 @ R^T
    gemm_f32_wmma<true><<<ggrid, gblock, 0, stream>>>(T2, rot, y);
}